// ButterflyLinear_33835752358558
// MI455X (gfx1250) — compile-verified
//
#include <hip/hip_runtime.h>

typedef __attribute__((ext_vector_type(2))) float v2f;
typedef __attribute__((ext_vector_type(8))) float v8f;

#define IN_N   4096
#define M_ROWS 16384
#define NPAIRS 2048
#define NFACT  12

// ---------------------------------------------------------------------------
// Kernel 1: fold the 12 per-pair 2x2 factor matrices into one 2x2 per pair.
// The reference applies the SAME adjacent-pair reshape every stage, so the
// stages compose: G[n] = F0[n] * F1[n] * ... * F11[n], then scale by alpha.
// Output G: [2048][2][2] f32 (32 KB) in workspace -> stays hot in L2.
// ---------------------------------------------------------------------------
__global__ void __launch_bounds__(256)
fold_factors_kernel(const float* __restrict__ factors,
                    const float* __restrict__ alpha,
                    float* __restrict__ G) {
    int n = blockIdx.x * blockDim.x + threadIdx.x;
    if (n >= NPAIRS) return;
    const float* F0 = factors + (size_t)n * 4;
    float g00 = F0[0], g01 = F0[1], g10 = F0[2], g11 = F0[3];
    #pragma unroll
    for (int f = 1; f < NFACT; ++f) {
        const float* Ff = factors + ((size_t)f * NPAIRS + n) * 4;
        float f00 = Ff[0], f01 = Ff[1], f10 = Ff[2], f11 = Ff[3];
        float h00 = g00 * f00 + g01 * f10;
        float h01 = g00 * f01 + g01 * f11;
        float h10 = g10 * f00 + g11 * f10;
        float h11 = g10 * f01 + g11 * f11;
        g00 = h00; g01 = h01; g10 = h10; g11 = h11;
    }
    float a = alpha[0];
    float* Gp = G + (size_t)n * 4;
    Gp[0] = g00 * a;
    Gp[1] = g01 * a;
    Gp[2] = g10 * a;
    Gp[3] = g11 * a;
}

// ---------------------------------------------------------------------------
// Kernel 2: streaming pass  Y = X * blockdiag(G)  using V_WMMA_F32_16X16X4_F32.
// One wave computes a 16x16 output tile as 4 accumulating K=4 WMMAs.
// B chunks (from G) depend only on the tile column -> built once per wave and
// reused across 4 M-tiles. Memory-bound: 512 MB total traffic (~22us floor
// at 23.3 TB/s); WMMA FLOPs are ~1000x below the compute roofline.
//
// VGPR layouts (wave32, per ISA 7.12.2):
//   A (16x4):  lane L, vgpr v -> (M = L%16, K = 2*(L/16) + v)   [float2 load]
//   B (4x16):  lane L, vgpr v -> (K = v + 2*(L/16), N = L%16)
//   D (16x16): lane L, vgpr r -> (M = r + 8*(L/16), N = L%16)
// Uniform control flow everywhere -> EXEC all ones at each WMMA.
// ---------------------------------------------------------------------------
__global__ void __launch_bounds__(256)
butterfly_apply_wmma_kernel(const float* __restrict__ X,
                            const float* __restrict__ G,
                            float* __restrict__ Y) {
    const int lane  = threadIdx.x & 31;
    const int wave  = threadIdx.x >> 5;
    const int half  = lane >> 4;      // 0 or 1
    const int l16   = lane & 15;
    const int tileN = blockIdx.x;                 // 0..255
    const int colBase = tileN * 16;
    const int mgroup  = blockIdx.y * 8 + wave;    // 0..255

    // Build the 4 B chunks (4x16 each, 2 VGPRs) from folded G.
    // Chunk t covers X columns [4t, 4t+4) of this 16-col tile:
    //   pair-in-chunk j == half, b == v, c = l16 - 4t - 2*half.
    v2f bmat[4];
    #pragma unroll
    for (int t = 0; t < 4; ++t) {
        int c    = l16 - 4 * t - 2 * half;
        int pair = tileN * 8 + 2 * t + half;
        float b0 = 0.0f, b1 = 0.0f;
        if (c == 0 || c == 1) {
            const float* Gp = G + (size_t)pair * 4;
            b0 = Gp[0 * 2 + c];   // b = 0
            b1 = Gp[1 * 2 + c];   // b = 1
        }
        bmat[t].x = b0;
        bmat[t].y = b1;
    }

    #pragma unroll
    for (int i = 0; i < 4; ++i) {
        const int tileM   = mgroup * 4 + i;       // 0..1023
        const int rowBase = tileM * 16;

        // Prefetch next M-tile's slice of X while this tile computes.
        if (i < 3) {
            const float* pn = X + (size_t)(rowBase + 16 + l16) * IN_N
                                + colBase + 2 * half;
            __builtin_prefetch(pn, 0, 0);
        }

        // Load A chunks: lane holds (M = l16, K = 2*half + {0,1}) as float2.
        v2f amat[4];
        #pragma unroll
        for (int t = 0; t < 4; ++t) {
            const float* p = X + (size_t)(rowBase + l16) * IN_N
                               + colBase + 4 * t + 2 * half;
            amat[t] = *reinterpret_cast<const v2f*>(p);
        }

        // D = sum_t A_t * B_t  (accumulating WMMA chain).
        v8f acc = {};
        #pragma unroll
        for (int t = 0; t < 4; ++t) {
            acc = __builtin_amdgcn_wmma_f32_16x16x4_f32(
                /*neg_a=*/false, amat[t],
                /*neg_b=*/false, bmat[t],
                /*c_mod=*/(short)0, acc,
                /*reuse_a=*/false, /*reuse_b=*/false);
        }

        // Store D: vgpr r, lane -> row rowBase + r + 8*half, col colBase + l16.
        #pragma unroll
        for (int r = 0; r < 8; ++r) {
            Y[(size_t)(rowBase + r + 8 * half) * IN_N + colBase + l16] = acc[r];
        }
    }
}

extern "C" void kernel_launch(void* const* d_in, const int* in_sizes, int n_in,
                              void* d_out, int out_size, void* d_ws, size_t ws_size,
                              hipStream_t stream) {
    const float* x       = (const float*)d_in[0];   // [4,4096,4096] f32
    const float* factors = (const float*)d_in[1];   // [12,2048,2,2] f32
    const float* alpha   = (const float*)d_in[2];   // [1] f32
    float*       out     = (float*)d_out;           // [4,4096,4096] f32
    float*       G       = (float*)d_ws;            // [2048,2,2] f32 (32 KB)

    // Fold 12 factor stages (+alpha) into one 2x2 per pair.
    fold_factors_kernel<<<dim3(8), dim3(256), 0, stream>>>(factors, alpha, G);

    // Single streaming pass: 256 tile-columns x (32 blocks * 8 waves * 4 tiles)
    // = 256 x 1024 sixteen-by-sixteen output tiles.
    butterfly_apply_wmma_kernel<<<dim3(256, 32), dim3(256), 0, stream>>>(x, G, out);
}